// SEIRCoupledModel_46256797778321
// MI455X (gfx1250) — compile-verified
//
#include <hip/hip_runtime.h>
#include <math.h>

// CDNA5 / gfx1250, wave32.
typedef __attribute__((ext_vector_type(2))) float v2f;
typedef __attribute__((ext_vector_type(8))) float v8f;

#define T_MONTHS 240
#define NTILES   15            // 240 / 16 rows per WMMA tile
#define SIGMA_H  (1.0f / 5.9f)
#define GAMMA_H  (1.0f / 14.0f)
#define N_H      14260000.0f

__device__ __forceinline__ float softplusf(float x) {
    // numerically stable softplus = max(x,0) + log1p(exp(-|x|))
    return fmaxf(x, 0.0f) + log1pf(expf(-fabsf(x)));
}

__global__ void __launch_bounds__(32)
seir_mlp_scan_kernel(const float* __restrict__ wn,     // [240*3] weather_norm
                     const float* __restrict__ mnorm,  // [240]   m_norm
                     const float* __restrict__ W1,     // [3*16]  (row-major [in,out])
                     const float* __restrict__ b1,     // [16]
                     const float* __restrict__ W2,     // [16*16]
                     const float* __restrict__ b2,     // [16]
                     const float* __restrict__ W3,     // [16*1]
                     const float* __restrict__ b3,     // [1]
                     const float* __restrict__ lbs,    // [1] log_beta_scale
                     const float* __restrict__ leta,   // [1] log_eta
                     const int*   __restrict__ days_p, // [1]
                     float* __restrict__ out)          // [480]: cases[240] ++ beta[240]
{
    __shared__ float sh_h [256];   // layer-1 activations, row-major [m][k]
    __shared__ float sh_h2[256];   // layer-2 activations, row-major [m][k]
    __shared__ float sh_beta[T_MONTHS];

    const int lane  = threadIdx.x;     // 0..31 (wave32)
    const int nloc  = lane & 15;       // N (or M) slot within half-wave
    const int khalf = lane >> 4;       // 0: K={0,1}; 1: K={2,3}  (A/B 32-bit layout)

    // ---- per-lane constant B operands (weights), loaded once --------------
    // B 4x16 layout: VGPR0 = rows K=0 (lanes 0-15), K=2 (lanes 16-31); VGPR1 = K=1 / K=3.
    v2f B1;
    B1.x = W1[(2 * khalf) * 16 + nloc];            // K=0 or K=2
    B1.y = khalf ? 0.0f : W1[1 * 16 + nloc];       // K=1, K=3 is zero padding (K: 3 -> 4)
    v2f B2c[4];
    #pragma unroll
    for (int kc = 0; kc < 4; ++kc) {
        const int kb = 4 * kc + 2 * khalf;
        B2c[kc].x = W2[kb * 16 + nloc];
        B2c[kc].y = W2[(kb + 1) * 16 + nloc];
    }
    const float bias1 = b1[nloc];
    const float bias2 = b2[nloc];
    const float bias3 = b3[0];
    const float beta_scale = expf(lbs[0]);
    float w3r[16];
    #pragma unroll
    for (int k = 0; k < 16; ++k) w3r[k] = W3[k];

    // ---- phase 1: MLP beta(t) for all 240 timesteps, 15 tiles of 16 rows --
    for (int t = 0; t < NTILES; ++t) {
        // layer 1: [16x3] @ [3x16] + b1, via one V_WMMA_F32_16X16X4_F32 (K padded to 4)
        const int row = t * 16 + nloc;             // A 16x4: lane holds row M=nloc
        v2f A1;
        A1.x = wn[row * 3 + 2 * khalf];            // K=0 or K=2
        A1.y = khalf ? 0.0f : wn[row * 3 + 1];     // K=1, or K=3 zero pad
        v8f acc1;
        #pragma unroll
        for (int r = 0; r < 8; ++r) acc1[r] = bias1;   // C = bias (broadcast over M)
        acc1 = __builtin_amdgcn_wmma_f32_16x16x4_f32(
            false, A1, false, B1, (short)0, acc1, false, false);
        #pragma unroll
        for (int r = 0; r < 8; ++r)                // D layout: VGPR r = rows r, r+8
            sh_h[(r + 8 * khalf) * 16 + nloc] = softplusf(acc1[r]);
        __syncthreads();

        // layer 2: [16x16] @ [16x16] + b2, via 4 chained K=4 WMMAs (D feeds C)
        v8f acc2;
        #pragma unroll
        for (int r = 0; r < 8; ++r) acc2[r] = bias2;
        #pragma unroll
        for (int kc = 0; kc < 4; ++kc) {
            const int kb = 4 * kc + 2 * khalf;
            v2f A2;
            A2.x = sh_h[nloc * 16 + kb];
            A2.y = sh_h[nloc * 16 + kb + 1];
            acc2 = __builtin_amdgcn_wmma_f32_16x16x4_f32(
                false, A2, false, B2c[kc], (short)0, acc2, false, false);
        }
        #pragma unroll
        for (int r = 0; r < 8; ++r)
            sh_h2[(r + 8 * khalf) * 16 + nloc] = softplusf(acc2[r]);
        __syncthreads();

        // layer 3: [16x16] @ [16x1] + b3, sigmoid, scale. One row per lane (lanes 0-15).
        if (khalf == 0) {
            float acc = bias3;
            #pragma unroll
            for (int k = 0; k < 16; ++k) acc += sh_h2[nloc * 16 + k] * w3r[k];
            const float bnorm = 1.0f / (1.0f + expf(-acc));
            const float beta  = bnorm * beta_scale;
            sh_beta[t * 16 + nloc] = beta;
            out[T_MONTHS + t * 16 + nloc] = beta;       // second tuple element
        }
        __syncthreads();   // protect sh_h/sh_h2 reuse next tile
    }

    // ---- phase 2: strictly sequential SEIR daily scan (one lane) ----------
    // 7200 dependent iterations; no parallelism exists. R is write-only in the
    // reference and dropped; I_pos only fed R and is dropped too.
    if (lane == 0) {
        const float eta  = expf(leta[0]);
        const int   days = days_p[0];
        float S = N_H - 1.0f, E = 0.0f, I = 1.0f;
        for (int t = 0; t < T_MONTHS; ++t) {
            const float coef = sh_beta[t] * mnorm[t] * (1.0f / N_H); // b*m/n_h hoisted
            float month_new = 0.0f;
            for (int d = 0; d < days; ++d) {
                const float lam = fminf(coef * I, 0.01f);
                const float ne  = lam * S;
                const float Ep  = fmaxf(E, 0.0f);
                const float Sn  = fmaxf(S - ne, 0.0f);
                const float En  = fmaxf(E + ne + eta - SIGMA_H * E, 0.0f);
                const float In  = fmaxf(I + SIGMA_H * Ep - GAMMA_H * I, 0.0f);
                month_new += SIGMA_H * Ep;
                S = Sn; E = En; I = In;
            }
            out[t] = month_new;                        // first tuple element
        }
    }
}

extern "C" void kernel_launch(void* const* d_in, const int* in_sizes, int n_in,
                              void* d_out, int out_size, void* d_ws, size_t ws_size,
                              hipStream_t stream) {
    (void)in_sizes; (void)n_in; (void)out_size; (void)d_ws; (void)ws_size;
    seir_mlp_scan_kernel<<<1, 32, 0, stream>>>(
        (const float*)d_in[0],   // weather_norm [240,3]
        (const float*)d_in[1],   // m_norm [240]
        (const float*)d_in[2],   // W1 [3,16]
        (const float*)d_in[3],   // b1 [16]
        (const float*)d_in[4],   // W2 [16,16]
        (const float*)d_in[5],   // b2 [16]
        (const float*)d_in[6],   // W3 [16,1]
        (const float*)d_in[7],   // b3 [1]
        (const float*)d_in[8],   // log_beta_scale
        (const float*)d_in[9],   // log_eta
        (const int*)  d_in[10],  // days
        (float*)d_out);          // cases[240] ++ beta[240]
}